// GCN_14250701488874
// MI455X (gfx1250) — compile-verified
//
#include <hip/hip_runtime.h>
#include <cmath>

// ---------------------------------------------------------------------------
// GCN 3-layer forward for MI455X (gfx1250), wave32.
//   layer: h = x @ W^T + b ; agg = segment_sum(h[dst]*ew, src) ; ELU (layers 1,2)
// GEMM: V_WMMA_F32_16X16X4_F32, one wave owns a 16 x N stripe (N-register-
//       blocked: X read exactly once, 8 independent WMMAs per K-step).
// SpMM: edge-parallel, float4 gather + native global_atomic_add_f32 (DEV scope).
// ---------------------------------------------------------------------------

typedef float v2f __attribute__((ext_vector_type(2)));
typedef float v8f __attribute__((ext_vector_type(8)));

// Native no-return f32 atomic add at device scope (L2 atomic units).
// STOREcnt-tracked; s_endpgm's implicit wait-idle covers completion.
__device__ __forceinline__ void atomic_add_f32_dev(float* p, float v)
{
    asm volatile("global_atomic_add_f32 %0, %1, off scope:SCOPE_DEV"
                 :: "v"(p), "v"(v) : "memory");
}

// ---------------------------------------------------------------------------
// Dense GEMM: H[M,N] = X[M,K] @ W[N,K]^T + bias[N]
// One wave computes a 16 x N stripe (NT = N/16 accumulators), stepping K by 4
// with V_WMMA_F32_16X16X4_F32.
//
// A-fragment (16x4 f32, 2 VGPRs/lane):
//   lanes 0-15 : row M = lane,    {K=k0,   K=k0+1}
//   lanes 16-31: row M = lane-16, {K=k0+2, K=k0+3}
// B-fragment (4x16 f32, B = W^T so B[k][n] = W[n][k]) mirrored:
//   lanes 0-15 : col N = lane,    {K=k0,   K=k0+1}
//   lanes 16-31: col N = lane-16, {K=k0+2, K=k0+3}
// Both are contiguous float2 loads from row-major X / W.
// C/D (8 VGPRs): vgpr i -> row M = i + 8*(lane>=16), col N = lane&15.
// ---------------------------------------------------------------------------
template <int K, int N>
__global__ __launch_bounds__(256) void gcn_gemm_wmma(
    const float* __restrict__ X, const float* __restrict__ W,
    const float* __restrict__ bias, float* __restrict__ H, int M)
{
    constexpr int NT = N / 16;
    const int wave = (blockIdx.x * blockDim.x + threadIdx.x) >> 5;
    const int lane = threadIdx.x & 31;
    if (wave >= M / 16) return;          // M = 100000 is a multiple of 16

    const int tm = wave * 16;
    const int r  = lane & 15;            // row of A-tile / col of B-tile
    const int kh = (lane >> 4) * 2;      // K sub-offset within the 4-wide step

    const float* xrow = X + (size_t)(tm + r) * K + kh;
    const float* wrow = W + (size_t)r * K + kh;

    v8f acc[NT];
#pragma unroll
    for (int t = 0; t < NT; ++t) acc[t] = (v8f)0.0f;

    for (int k0 = 0; k0 < K; k0 += 4) {
        v2f a = *(const v2f*)(xrow + k0);
#pragma unroll
        for (int t = 0; t < NT; ++t) {   // NT independent WMMAs share the A frag
            v2f b = *(const v2f*)(wrow + (size_t)t * 16 * K + k0);
            // (neg_a, A, neg_b, B, c_mod, C, reuse_a, reuse_b)
            acc[t] = __builtin_amdgcn_wmma_f32_16x16x4_f32(
                false, a, false, b, (short)0, acc[t], false, false);
        }
    }

    const int row0 = tm + ((lane >> 4) << 3);   // +8 for upper half-wave
#pragma unroll
    for (int t = 0; t < NT; ++t) {
        const int col = t * 16 + r;
        const float bv = bias[col];
#pragma unroll
        for (int i = 0; i < 8; ++i)
            H[(size_t)(row0 + i) * N + col] = acc[t][i] + bv;
    }
}

// ---------------------------------------------------------------------------
// Edge-parallel SpMM scatter: AGG[src[e], :] += H[dst[e], :] * ew[e]
// One wave per edge; lane covers VPT consecutive features (F = 32*VPT).
// float4/float2 gather (global_load_b128/b64) + native f32 L2 atomics.
// ---------------------------------------------------------------------------
template <int VPT>
__global__ __launch_bounds__(256) void gcn_spmm_atomic(
    const float* __restrict__ H, const int* __restrict__ src,
    const int* __restrict__ dst, const float* __restrict__ ew,
    float* __restrict__ AGG, int nE)
{
    const int e = (blockIdx.x * blockDim.x + threadIdx.x) >> 5;
    if (e >= nE) return;
    const int lane = threadIdx.x & 31;
    constexpr int F = 32 * VPT;

    const int s = src[e];
    const int d = dst[e];
    const float w = ew[e];

    const float* hp = H   + (size_t)d * F + lane * VPT;
    float*       ap = AGG + (size_t)s * F + lane * VPT;

    if constexpr (VPT == 4) {
        float4 v = *(const float4*)hp;
        atomic_add_f32_dev(ap + 0, v.x * w);
        atomic_add_f32_dev(ap + 1, v.y * w);
        atomic_add_f32_dev(ap + 2, v.z * w);
        atomic_add_f32_dev(ap + 3, v.w * w);
    } else {
        float2 v = *(const float2*)hp;
        atomic_add_f32_dev(ap + 0, v.x * w);
        atomic_add_f32_dev(ap + 1, v.y * w);
    }
}

// ---------------------------------------------------------------------------
// Elementwise ELU (alpha = 1): x > 0 ? x : exp(x)-1 ; float4-vectorized.
// ---------------------------------------------------------------------------
__global__ __launch_bounds__(256) void gcn_elu4(float4* __restrict__ A, long n4)
{
    long i = (long)blockIdx.x * blockDim.x + threadIdx.x;
    if (i < n4) {
        float4 v = A[i];
        v.x = v.x > 0.0f ? v.x : expm1f(v.x);
        v.y = v.y > 0.0f ? v.y : expm1f(v.y);
        v.z = v.z > 0.0f ? v.z : expm1f(v.z);
        v.w = v.w > 0.0f ? v.w : expm1f(v.w);
        A[i] = v;
    }
}

// ---------------------------------------------------------------------------
// Orchestration. Workspace: two ping-pong f32 buffers of M*128 each
// (51.2 MB each -> both resident in the 192 MB L2).
// ---------------------------------------------------------------------------
extern "C" void kernel_launch(void* const* d_in, const int* in_sizes, int n_in,
                              void* d_out, int out_size, void* d_ws, size_t ws_size,
                              hipStream_t stream)
{
    const float* x   = (const float*)d_in[0];
    const int*   src = (const int*)  d_in[1];
    const int*   dst = (const int*)  d_in[2];
    const float* ew  = (const float*)d_in[3];
    const float* W1  = (const float*)d_in[4];
    const float* b1  = (const float*)d_in[5];
    const float* W2  = (const float*)d_in[6];
    const float* b2  = (const float*)d_in[7];
    const float* W3  = (const float*)d_in[8];
    const float* b3  = (const float*)d_in[9];
    float* out = (float*)d_out;

    const int M  = in_sizes[0] / 256;   // 100000 nodes
    const int nE = in_sizes[1];         // 1.6M edges

    float* bufA = (float*)d_ws;                  // GEMM output h
    float* bufB = bufA + (size_t)M * 128;        // SpMM output agg

    const dim3 blk(256);
    const dim3 gGemm((M / 16 + 7) / 8);          // 1 wave per 16-row stripe
    const dim3 gSpmm((nE + 7) / 8);              // 8 waves/block, 1 edge/wave
    const long n4 = (long)M * 128 / 4;
    const dim3 gElu((unsigned)((n4 + 255) / 256));

    // ---- Layer 1: 256 -> 128, ELU --------------------------------------
    gcn_gemm_wmma<256, 128><<<gGemm, blk, 0, stream>>>(x, W1, b1, bufA, M);
    hipMemsetAsync(bufB, 0, (size_t)M * 128 * sizeof(float), stream);
    gcn_spmm_atomic<4><<<gSpmm, blk, 0, stream>>>(bufA, src, dst, ew, bufB, nE);
    gcn_elu4<<<gElu, blk, 0, stream>>>((float4*)bufB, n4);

    // ---- Layer 2: 128 -> 128, ELU --------------------------------------
    gcn_gemm_wmma<128, 128><<<gGemm, blk, 0, stream>>>(bufB, W2, b2, bufA, M);
    hipMemsetAsync(bufB, 0, (size_t)M * 128 * sizeof(float), stream);   // bufB consumed above
    gcn_spmm_atomic<4><<<gSpmm, blk, 0, stream>>>(bufA, src, dst, ew, bufB, nE);
    gcn_elu4<<<gElu, blk, 0, stream>>>((float4*)bufB, n4);

    // ---- Layer 3: 128 -> 64, no activation, aggregate into d_out -------
    gcn_gemm_wmma<128, 64><<<gGemm, blk, 0, stream>>>(bufB, W3, b3, bufA, M);
    hipMemsetAsync(out, 0, (size_t)M * 64 * sizeof(float), stream);
    gcn_spmm_atomic<2><<<gSpmm, blk, 0, stream>>>(bufA, src, dst, ew, out, nE);
}